// FeatureRecalibrationLoss_25872882991969
// MI455X (gfx1250) — compile-verified
//
#include <hip/hip_runtime.h>
#include <hip/hip_bf16.h>
#include <math.h>

typedef float v2f __attribute__((ext_vector_type(2)));
typedef float v8f __attribute__((ext_vector_type(8)));

#define NUM_CLASSES 7
#define DIM 512
#define BLOCK 256
#define NBLOCKS 512

// Workspace layout (floats):
//   [0 .. 3584)            S[7][512] per-class feature sums
//   [3584]                 ce_sum
//   [3585]                 fsq_sum (sum of squares of all features)
//   [3586]                 l1_sum
//   [3587 .. 3594)         counts[7] (accumulated as float; exact up to 2^24)
#define WS_S     0
#define WS_CE    (NUM_CLASSES * DIM)
#define WS_FSQ   (WS_CE + 1)
#define WS_L1    (WS_CE + 2)
#define WS_CNT   (WS_CE + 3)
#define WS_TOTAL (WS_CNT + NUM_CLASSES)

__global__ void frl_init_ws(float* __restrict__ ws) {
    int i = blockIdx.x * blockDim.x + threadIdx.x;
    if (i < WS_TOTAL) ws[i] = 0.0f;
}

__global__ __launch_bounds__(BLOCK) void frl_main(
    const float* __restrict__ outputs,       // [B, 7]
    const float* __restrict__ features,      // [B, 512]
    const long long* __restrict__ targets,   // [B] int64
    const float* __restrict__ aw,            // [B, 512]
    float* __restrict__ ws, int B)
{
    __shared__ float red[BLOCK];
    __shared__ int   cls_cnt[NUM_CLASSES];

    const int tid  = threadIdx.x;
    const int lane = tid & 31;
    const int wave = tid >> 5;                      // 0..7, owns dims [wave*64, wave*64+64)
    const int rowsPerBlock = B / gridDim.x;         // 128 for B=65536, 512 blocks
    const int row0 = blockIdx.x * rowsPerBlock;

    // ---------------- WMMA one-hot segment-sum + sum-of-squares ----------------
    // A (16x4 f32): lane = M(class 0..15), VGPR0 = K(0 | 2), VGPR1 = K(1 | 3)
    // B (4x16 f32): lane = N(dim),        VGPR0 = K(0 | 2), VGPR1 = K(1 | 3)
    // D (16x16 f32): VGPR r, lane L -> S[class = r + 8*(L>>4)][dim = nbase+tile*16+(L&15)]
    const int half  = lane >> 4;                    // which K half this lane covers
    const int m     = lane & 15;
    const int nbase = wave * 64;

    v8f acc0 = {}, acc1 = {}, acc2 = {}, acc3 = {};
    float fsq = 0.0f;

    for (int k0 = row0; k0 < row0 + rowsPerBlock; k0 += 4) {
        const int rA = k0 + 2 * half;               // rows this lane covers (K = 2*half, 2*half+1)
        long long tA = targets[rA + 0];
        long long tB = targets[rA + 1];
        v2f a;
        a.x = (tA == (long long)m) ? 1.0f : 0.0f;
        a.y = (tB == (long long)m) ? 1.0f : 0.0f;

        const float* r0p = features + (size_t)(rA + 0) * DIM + nbase + m;
        const float* r1p = features + (size_t)(rA + 1) * DIM + nbase + m;

        // prefetch the feature rows ~4 K-steps ahead (emits global_prefetch_b8)
        __builtin_prefetch(features + (size_t)(rA + 16) * DIM + nbase + m, 0, 0);

        v2f b0, b1, b2, b3;
        b0.x = r0p[0];  b0.y = r1p[0];
        b1.x = r0p[16]; b1.y = r1p[16];
        b2.x = r0p[32]; b2.y = r1p[32];
        b3.x = r0p[48]; b3.y = r1p[48];

        fsq += b0.x*b0.x + b0.y*b0.y + b1.x*b1.x + b1.y*b1.y
             + b2.x*b2.x + b2.y*b2.y + b3.x*b3.x + b3.y*b3.y;

        acc0 = __builtin_amdgcn_wmma_f32_16x16x4_f32(false, a, false, b0, (short)0, acc0, false, false);
        acc1 = __builtin_amdgcn_wmma_f32_16x16x4_f32(false, a, false, b1, (short)0, acc1, false, false);
        acc2 = __builtin_amdgcn_wmma_f32_16x16x4_f32(false, a, false, b2, (short)0, acc2, false, false);
        acc3 = __builtin_amdgcn_wmma_f32_16x16x4_f32(false, a, false, b3, (short)0, acc3, false, false);
    }

    // Merge this block's S partials. Valid classes live in lanes 0-15 (half==0), rows r=0..6.
    if (half == 0) {
        #pragma unroll
        for (int r = 0; r < NUM_CLASSES; ++r) {
            atomicAdd(&ws[WS_S + r * DIM + nbase +  0 + m], acc0[r]);
            atomicAdd(&ws[WS_S + r * DIM + nbase + 16 + m], acc1[r]);
            atomicAdd(&ws[WS_S + r * DIM + nbase + 32 + m], acc2[r]);
            atomicAdd(&ws[WS_S + r * DIM + nbase + 48 + m], acc3[r]);
        }
    }

    // ---------------- Cross-entropy + class counts (one row per thread) --------
    if (tid < NUM_CLASSES) cls_cnt[tid] = 0;
    __syncthreads();

    float ce = 0.0f;
    for (int i = row0 + tid; i < row0 + rowsPerBlock; i += BLOCK) {
        const float* o = outputs + (size_t)i * NUM_CLASSES;
        float x0 = o[0], x1 = o[1], x2 = o[2], x3 = o[3], x4 = o[4], x5 = o[5], x6 = o[6];
        float mx = fmaxf(fmaxf(fmaxf(x0, x1), fmaxf(x2, x3)),
                         fmaxf(fmaxf(x4, x5), x6));
        float s = expf(x0 - mx) + expf(x1 - mx) + expf(x2 - mx) + expf(x3 - mx)
                + expf(x4 - mx) + expf(x5 - mx) + expf(x6 - mx);
        float lse = mx + logf(s);
        int t = (int)targets[i];
        ce += lse - o[t];
        atomicAdd(&cls_cnt[t], 1);
    }

    // ---------------- L1 over attention weights (vectorized grid-stride) -------
    float l1 = 0.0f;
    {
        const float4* aw4 = (const float4*)aw;
        size_t total4  = ((size_t)B * DIM) >> 2;            // B*DIM divisible by 4
        size_t stride4 = (size_t)gridDim.x * BLOCK;
        for (size_t j = (size_t)blockIdx.x * BLOCK + tid; j < total4; j += stride4) {
            float4 v = aw4[j];
            l1 += fabsf(v.x) + fabsf(v.y) + fabsf(v.z) + fabsf(v.w);
        }
    }

    // ---------------- Block reductions -> global atomics ------------------------
    __syncthreads();
    red[tid] = fsq; __syncthreads();
    for (int s = BLOCK / 2; s > 0; s >>= 1) { if (tid < s) red[tid] += red[tid + s]; __syncthreads(); }
    if (tid == 0) atomicAdd(&ws[WS_FSQ], red[0]);
    __syncthreads();

    red[tid] = ce; __syncthreads();
    for (int s = BLOCK / 2; s > 0; s >>= 1) { if (tid < s) red[tid] += red[tid + s]; __syncthreads(); }
    if (tid == 0) atomicAdd(&ws[WS_CE], red[0]);
    __syncthreads();

    red[tid] = l1; __syncthreads();
    for (int s = BLOCK / 2; s > 0; s >>= 1) { if (tid < s) red[tid] += red[tid + s]; __syncthreads(); }
    if (tid == 0) atomicAdd(&ws[WS_L1], red[0]);
    __syncthreads();

    if (tid < NUM_CLASSES) atomicAdd(&ws[WS_CNT + tid], (float)cls_cnt[tid]);
}

__global__ __launch_bounds__(256) void frl_finalize(const float* __restrict__ ws,
                                                    float* __restrict__ out, int B)
{
    __shared__ float red[256];
    const int tid = threadIdx.x;

    // centers_norm = sum_c ||s_c||^2 / max(n_c, 1)   (absent class -> s_c = 0)
    float cn = 0.0f;
    for (int idx = tid; idx < NUM_CLASSES * DIM; idx += 256) {
        int   c = idx / DIM;
        float n = fmaxf(ws[WS_CNT + c], 1.0f);
        float s = ws[WS_S + idx];
        cn += (s * s) / n;
    }
    red[tid] = cn; __syncthreads();
    for (int s = 128; s > 0; s >>= 1) { if (tid < s) red[tid] += red[tid + s]; __syncthreads(); }

    if (tid == 0) {
        float invB    = 1.0f / (float)B;
        float ce      = ws[WS_CE] * invB;
        float center  = (ws[WS_FSQ] - red[0]) * invB;   // sum_i ||f_i - c_{t_i}||^2 / B
        float l1      = ws[WS_L1] / ((float)B * (float)DIM);
        out[0] = ce + 0.1f * center + 0.01f * l1;       // total_loss
        out[1] = ce;
        out[2] = center;
        out[3] = l1;
    }
}

extern "C" void kernel_launch(void* const* d_in, const int* in_sizes, int n_in,
                              void* d_out, int out_size, void* d_ws, size_t ws_size,
                              hipStream_t stream) {
    const float*     outputs  = (const float*)d_in[0];
    const float*     features = (const float*)d_in[1];
    const long long* targets  = (const long long*)d_in[2];  // int64 in reference
    const float*     aw       = (const float*)d_in[3];
    float* ws  = (float*)d_ws;
    float* out = (float*)d_out;

    const int B = in_sizes[0] / NUM_CLASSES;   // 65536

    frl_init_ws<<<(WS_TOTAL + 255) / 256, 256, 0, stream>>>(ws);
    frl_main<<<NBLOCKS, BLOCK, 0, stream>>>(outputs, features, targets, aw, ws, B);
    frl_finalize<<<1, 256, 0, stream>>>(ws, out, B);
}